// RGCN_70368744178402
// MI455X (gfx1250) — compile-verified
//
#include <hip/hip_runtime.h>
#include <hip/hip_bf16.h>

// Problem constants (from the reference)
#define NN    50000       // nodes
#define RR    51          // relations
#define EE    16          // emb dim
#define CC    8           // classes
#define BB    40          // bases
#define NNZ   1600000     // edges
#define NE    (NN * EE)       // 800000  = columns of W1 GEMM
#define W1ELEMS ((size_t)RR * NE)  // 40,800,000 floats

typedef float v2f __attribute__((ext_vector_type(2)));
typedef float v8f __attribute__((ext_vector_type(8)));

// Guaranteed-hardware fp32 atomic add (global_atomic_add_f32, no return,
// STOREcnt-tracked) instead of a possible CAS loop.
__device__ __forceinline__ void atomAddF32(float* p, float v) {
  unsafeAtomicAdd(p, v);
}

// ---------------------------------------------------------------------------
// K0: init h = 0, out = bias2 broadcast (d_out is poisoned before timing)
// ---------------------------------------------------------------------------
__global__ void rgcn_init(float* __restrict__ h, float* __restrict__ out,
                          const float* __restrict__ bias2) {
  int i = blockIdx.x * blockDim.x + threadIdx.x;
  if (i < NE)        h[i] = 0.0f;
  if (i < NN * CC)   out[i] = bias2[i & (CC - 1)];
}

// ---------------------------------------------------------------------------
// K1: W2[r,e,c] = sum_b comps2[r,b] * bases2[b,e,c]     (51*128 outputs)
// ---------------------------------------------------------------------------
__global__ void rgcn_w2(const float* __restrict__ comps2,
                        const float* __restrict__ bases2,
                        float* __restrict__ W2) {
  int idx = blockIdx.x * blockDim.x + threadIdx.x;
  if (idx >= RR * EE * CC) return;
  int r  = idx >> 7;          // /128
  int ec = idx & 127;
  float acc = 0.0f;
#pragma unroll
  for (int b = 0; b < BB; ++b)
    acc += comps2[r * BB + b] * bases2[b * (EE * CC) + ec];
  W2[idx] = acc;
}

// ---------------------------------------------------------------------------
// K2: W1(51 x 800000) = comps1(51x40) @ bases1(40x800000), fp32 WMMA.
// One wave owns a 16-wide column tile; 4 M-tiles x 10 K-steps of
// v_wmma_f32_16x16x4_f32 per tile. A (comps1) preloaded in registers.
//
// bases1 is read exactly once -> non-temporal loads (keep L2 free for W1).
// W1 stores are regular-temporal: W1 (163 MB) + h (3.2 MB) fit in the
// 192 MB L2 and W1 is re-read randomly in the edge-gather phase.
//
// ISA layouts (wave32):
//   A 16x4:  lane%16 = M, half-wave selects K pair: v.x=K0+2h, v.y=K0+2h+1
//   B 4x16:  lane%16 = N column, half-wave selects row pair likewise
//   D 16x16: VGPR j: lanes0-15 row j, lanes16-31 row j+8, N = lane%16
// ---------------------------------------------------------------------------
__global__ void __launch_bounds__(256)
rgcn_w1_gemm(const float* __restrict__ comps1,
             const float* __restrict__ bases1,
             float* __restrict__ W1) {
  const int lane = threadIdx.x & 31;
  const int half = lane >> 4;        // 0 or 1
  const int l16  = lane & 15;

  // Preload all of A (comps1, zero-padded rows >= 51) into registers.
  v2f a[4][10];
#pragma unroll
  for (int mt = 0; mt < 4; ++mt) {
    int m = mt * 16 + l16;
#pragma unroll
    for (int ks = 0; ks < 10; ++ks) {
      int k = ks * 4 + half * 2;
      float ax = 0.0f, ay = 0.0f;
      if (m < RR) {
        ax = comps1[m * BB + k];
        ay = comps1[m * BB + k + 1];
      }
      a[mt][ks].x = ax;
      a[mt][ks].y = ay;
    }
  }

  const int wavesPerBlock = blockDim.x >> 5;
  const int waveId  = blockIdx.x * wavesPerBlock + (threadIdx.x >> 5);
  const int nWaves  = gridDim.x * wavesPerBlock;
  const int nTiles  = NE / 16;   // 50000

  for (int tile = waveId; tile < nTiles; tile += nWaves) {
    const int col = tile * 16 + l16;
    v8f acc[4] = {};
#pragma unroll
    for (int ks = 0; ks < 10; ++ks) {
      const int k = ks * 4 + half * 2;
      v2f b;
      b.x = __builtin_nontemporal_load(&bases1[(size_t)k * NE + col]);
      b.y = __builtin_nontemporal_load(&bases1[(size_t)(k + 1) * NE + col]);
#pragma unroll
      for (int mt = 0; mt < 4; ++mt) {
        acc[mt] = __builtin_amdgcn_wmma_f32_16x16x4_f32(
            /*neg_a=*/false, a[mt][ks], /*neg_b=*/false, b,
            /*c_mod=*/(short)0, acc[mt],
            /*reuse_a=*/false, /*reuse_b=*/false);
      }
    }
    // Store D tiles; only rows < 51 are valid. Regular-temporal (L2-resident).
#pragma unroll
    for (int mt = 0; mt < 4; ++mt) {
#pragma unroll
      for (int j = 0; j < 8; ++j) {
        int m = mt * 16 + j + half * 8;
        if (m < RR) W1[(size_t)m * NE + col] = acc[mt][j];
      }
    }
  }
}

// ---------------------------------------------------------------------------
// K3: h[hr[i]*16+e] += values[i] * W1[hc[i]*16+e]   (16 lanes per edge)
// W1 flat index: (p*N+o)*16 + e == hor_cols*16 + e  (layout matches K2 store)
// Edge streams are read-once -> NT loads; W1 gather regular (L2 hit path).
// ---------------------------------------------------------------------------
__global__ void rgcn_scatter_h(const float* __restrict__ values,
                               const int* __restrict__ hr,
                               const int* __restrict__ hc,
                               const float* __restrict__ W1,
                               float* __restrict__ h) {
  int t = blockIdx.x * blockDim.x + threadIdx.x;   // < NNZ*16 (25.6M)
  if (t >= NNZ * EE) return;
  int i = t >> 4;
  int e = t & 15;
  float v  = __builtin_nontemporal_load(&values[i]);
  int col  = __builtin_nontemporal_load(&hc[i]);
  int row  = __builtin_nontemporal_load(&hr[i]);
  float w  = W1[(size_t)col * EE + e];
  atomAddF32(&h[row * EE + e], v * w);
}

// ---------------------------------------------------------------------------
// K4: h = relu(h + bias1)
// ---------------------------------------------------------------------------
__global__ void rgcn_bias_relu(float* __restrict__ h,
                               const float* __restrict__ bias1) {
  int i = blockIdx.x * blockDim.x + threadIdx.x;
  if (i >= NE) return;
  float x = h[i] + bias1[i & 15];
  h[i] = x > 0.0f ? x : 0.0f;
}

// ---------------------------------------------------------------------------
// K5: fused layer 2:  out[s,c] += v * sum_e h[o,e] * W2[p,e,c]
// (p = ver_rows/N, s = ver_rows%N, o = ver_cols); 8 lanes per edge.
// W2 (51*128 floats = 26KB) staged in LDS; hv (163 MB) never materialized.
// ---------------------------------------------------------------------------
__global__ void rgcn_layer2(const float* __restrict__ values,
                            const int* __restrict__ vr,
                            const int* __restrict__ vc,
                            const float* __restrict__ h,
                            const float* __restrict__ W2,
                            float* __restrict__ out) {
  __shared__ float W2s[RR * EE * CC];   // 26112 bytes
  for (int j = threadIdx.x; j < RR * EE * CC; j += blockDim.x)
    W2s[j] = W2[j];
  __syncthreads();

  const long long total  = (long long)NNZ * CC;     // 12.8M
  const long long stride = (long long)gridDim.x * blockDim.x;
  for (long long t = (long long)blockIdx.x * blockDim.x + threadIdx.x;
       t < total; t += stride) {
    int i = (int)(t >> 3);
    int c = (int)(t & 7);
    int r   = __builtin_nontemporal_load(&vr[i]);
    int o   = __builtin_nontemporal_load(&vc[i]);
    float v = __builtin_nontemporal_load(&values[i]);
    int p = r / NN;              // constant-divisor -> mul-hi
    int s = r - p * NN;
    const float* hp = h + (size_t)o * EE;           // h is L2-resident (3.2MB)
    const float* wp = W2s + p * (EE * CC) + c;
    float acc = 0.0f;
#pragma unroll
    for (int e = 0; e < EE; ++e)
      acc += hp[e] * wp[e * CC];
    atomAddF32(&out[s * CC + c], v * acc);
  }
}

// ---------------------------------------------------------------------------
// Host launcher
// Input order: values, comps1, bases1, comps2, bases2, bias1, bias2,
//              hor_rows, hor_cols, ver_rows, ver_cols
// ---------------------------------------------------------------------------
extern "C" void kernel_launch(void* const* d_in, const int* in_sizes, int n_in,
                              void* d_out, int out_size, void* d_ws, size_t ws_size,
                              hipStream_t stream) {
  const float* values = (const float*)d_in[0];
  const float* comps1 = (const float*)d_in[1];
  const float* bases1 = (const float*)d_in[2];
  const float* comps2 = (const float*)d_in[3];
  const float* bases2 = (const float*)d_in[4];
  const float* bias1  = (const float*)d_in[5];
  const float* bias2  = (const float*)d_in[6];
  const int*   hr     = (const int*)d_in[7];
  const int*   hc     = (const int*)d_in[8];
  const int*   vr     = (const int*)d_in[9];
  const int*   vc     = (const int*)d_in[10];
  float* out = (float*)d_out;

  // Workspace layout
  float* W1 = (float*)d_ws;                 // 40.8M floats (163.2 MB)
  float* h  = W1 + W1ELEMS;                 // 800000 floats
  float* W2 = h + NE;                       // 6528 floats

  // K0: init h=0, out=bias2
  rgcn_init<<<(NE + 255) / 256, 256, 0, stream>>>(h, out, bias2);
  // K1: W2 basis composition
  rgcn_w2<<<(RR * EE * CC + 255) / 256, 256, 0, stream>>>(comps2, bases2, W2);
  // K2: W1 GEMM via WMMA (8 waves/block)
  rgcn_w1_gemm<<<832, 256, 0, stream>>>(comps1, bases1, W1);
  // K3: layer-1 sparse scatter (16 lanes per edge)
  {
    long long n = (long long)NNZ * EE;      // 25.6M
    rgcn_scatter_h<<<(int)((n + 255) / 256), 256, 0, stream>>>(values, hr, hc, W1, h);
  }
  // K4: bias + relu
  rgcn_bias_relu<<<(NE + 255) / 256, 256, 0, stream>>>(h, bias1);
  // K5: fused ver-graph scatter + relation einsum (8 lanes per edge)
  rgcn_layer2<<<4096, 256, 0, stream>>>(values, vr, vc, h, W2, out);
}